// Attention_55551107006804
// MI455X (gfx1250) — compile-verified
//
#include <hip/hip_runtime.h>
#include <math.h>
#include <stdint.h>

// ---------- problem constants (from reference) ----------
#define Hh   512          // H
#define Ff   2048         // H*SF
#define NB   16           // batch B
#define NS   2048         // seq S
#define BSr  (NB * NS)    // 32768 flattened (b,s) rows

typedef float v2f __attribute__((ext_vector_type(2)));
typedef float v8f __attribute__((ext_vector_type(8)));

__device__ __forceinline__ v8f wmma_f32(v2f a, v2f b, v8f c) {
  // D = A(16x4) * B(4x16) + C(16x16), fp32, wave32
  return __builtin_amdgcn_wmma_f32_16x16x4_f32(false, a, false, b, (short)0, c,
                                               false, false);
}

// CDNA5 async global->LDS copy (ASYNCcnt-tracked), GV addressing:
//   vdst = VGPR with LDS byte offset, vaddr = 64-bit global address.
__device__ __forceinline__ void async_ld_b128(uint32_t lds_off, const float* g) {
  asm volatile("global_load_async_to_lds_b128 %0, %1, off"
               :: "v"(lds_off), "v"((unsigned long long)(uintptr_t)g)
               : "memory");
}
__device__ __forceinline__ void wait_async0() {
  asm volatile("s_wait_asynccnt 0x0" ::: "memory");
}
__device__ __forceinline__ uint32_t lds_off_of(const void* p) {
  // generic LDS pointer = aperture | 32-bit LDS byte offset (ISA 10.2)
  return (uint32_t)(uintptr_t)p;
}

// =====================================================================
// k_h: h[b,j] = hidden[L-1][b,:] . Wh[j,:] + bh[j]      (16x512, tiny)
// =====================================================================
__global__ __launch_bounds__(256) void k_h(const float* __restrict__ hidden,
                                           const float* __restrict__ Wh,
                                           const float* __restrict__ bh,
                                           float* __restrict__ h) {
  const int o = blockIdx.x * 256 + threadIdx.x;   // 0..8191
  const int b = o >> 9, j = o & (Hh - 1);
  const float* hr = hidden + (size_t)1 * NB * Ff + (size_t)b * Ff;  // hidden[-1]
  const float* wr = Wh + (size_t)j * Ff;
  float acc = 0.f;
  for (int f = 0; f < Ff; f += 4) {
    float4 x = *(const float4*)(hr + f);
    float4 y = *(const float4*)(wr + f);
    acc += x.x * y.x + x.y * y.y + x.z * y.z + x.w * y.w;
  }
  h[o] = acc + bh[j];
}

// =====================================================================
// k_q: q[b,k] = h[b,:] . Wa[k, 0:H] + ba[k]             (16x512, tiny)
// =====================================================================
__global__ __launch_bounds__(256) void k_q(const float* __restrict__ h,
                                           const float* __restrict__ Wa,
                                           const float* __restrict__ ba,
                                           float* __restrict__ q) {
  const int o = blockIdx.x * 256 + threadIdx.x;
  const int b = o >> 9, k = o & (Hh - 1);
  const float* hr = h + (size_t)b * Hh;
  const float* wr = Wa + (size_t)k * (2 * Hh);    // Wa_h = Wa[:, :H]
  float acc = 0.f;
  for (int j = 0; j < Hh; j += 4) {
    float4 x = *(const float4*)(hr + j);
    float4 y = *(const float4*)(wr + j);
    acc += x.x * y.x + x.y * y.y + x.z * y.z + x.w * y.w;
  }
  q[o] = acc + ba[k];
}

// =====================================================================
// k_eo: eo[r, c] = enc[r, :] . We[c, :] + be[c]
//   M=32768, N=512, K=2048.  Block = 64 rows x 128 cols, 8 waves.
//   Async global->LDS B128 copies, double-buffered: loads of chunk k+1
//   overlap WMMA on chunk k; one barrier per K-iteration.
// =====================================================================
#define EO_ROWS 64
#define EO_COLS 128
#define EO_KC   32
#define EO_LD   (EO_KC + 4)   // stride 36 dwords: 16B-aligned rows, 64-bank clean

__global__ __launch_bounds__(256) void k_eo(const float* __restrict__ enc,
                                            const float* __restrict__ We,
                                            const float* __restrict__ be,
                                            float* __restrict__ eo) {
  __shared__ float sA[2][EO_ROWS * EO_LD];   // 2 x 9.0 KB
  __shared__ float sB[2][EO_COLS * EO_LD];   // 2 x 18.0 KB
  const int tid  = threadIdx.x;
  const int lane = tid & 31, w = tid >> 5;
  const int wr = w & 3, wc = w >> 2;         // 4 row strips x 2 col strips
  const int m  = lane & 15, lh = lane >> 4;
  const int r0 = blockIdx.x * EO_ROWS;
  const int c0 = blockIdx.y * EO_COLS;

  const uint32_t aBase[2] = { lds_off_of(&sA[0][0]), lds_off_of(&sA[1][0]) };
  const uint32_t bBase[2] = { lds_off_of(&sB[0][0]), lds_off_of(&sB[1][0]) };

  auto issue = [&](int buf, int k0) {
    // A tile: 64 rows x 32 f32 = 512 B128 xfers (2 per thread)
    for (int i = tid; i < EO_ROWS * (EO_KC / 4); i += 256) {
      const int row = i >> 3, kq = (i & 7) * 4;
      async_ld_b128(aBase[buf] + (uint32_t)(row * EO_LD + kq) * 4,
                    enc + (size_t)(r0 + row) * Ff + k0 + kq);
    }
    // B tile (We rows = output cols): 128 x 32 f32 = 1024 xfers (4 per thread)
    for (int i = tid; i < EO_COLS * (EO_KC / 4); i += 256) {
      const int row = i >> 3, kq = (i & 7) * 4;
      async_ld_b128(bBase[buf] + (uint32_t)(row * EO_LD + kq) * 4,
                    We + (size_t)(c0 + row) * Ff + k0 + kq);
    }
  };

  v8f acc[4];
  #pragma unroll
  for (int t = 0; t < 4; ++t)
    #pragma unroll
    for (int i = 0; i < 8; ++i) acc[t][i] = 0.f;

  const int NK = Ff / EO_KC;   // 64
  issue(0, 0);
  for (int kt = 0; kt < NK; ++kt) {
    const int buf = kt & 1;
    wait_async0();       // my async writes for `buf` have landed
    __syncthreads();     // everyone's landed; prev reads of buf^1 finished
    if (kt + 1 < NK) issue(buf ^ 1, (kt + 1) * EO_KC);

    const float* A  = &sA[buf][0];
    const float* Bt = &sB[buf][0];
    #pragma unroll
    for (int kk = 0; kk < EO_KC; kk += 4) {
      const int ka = kk + lh * 2;        // lanes 0-15: K0,K1; lanes 16-31: K2,K3
      v2f a;
      a.x = A[(wr * 16 + m) * EO_LD + ka];
      a.y = A[(wr * 16 + m) * EO_LD + ka + 1];
      #pragma unroll
      for (int t = 0; t < 4; ++t) {
        v2f b;
        const int col = wc * 64 + t * 16 + m;
        b.x = Bt[col * EO_LD + ka];
        b.y = Bt[col * EO_LD + ka + 1];
        acc[t] = wmma_f32(a, b, acc[t]);
      }
    }
  }
  // epilogue: add bias, store (C layout: lane half selects M=i / M=i+8)
  #pragma unroll
  for (int t = 0; t < 4; ++t) {
    const int col  = c0 + wc * 64 + t * 16 + m;
    const float bv = be[col];
    #pragma unroll
    for (int i = 0; i < 8; ++i) {
      const int row = r0 + wr * 16 + lh * 8 + i;
      eo[(size_t)row * Hh + col] = acc[t][i] + bv;
    }
  }
}

// =====================================================================
// k_energy: scores[r] = sum_k tanh( eo[r,:] . Wa_e[k,:] + q[b,k] ) * v[k]
//   M=32768, N=512 (k), K=512 (h).  Block = 64 rows, 8 waves,
//   N tiled by 32 (1 accum tile/wave), K streamed in 128-chunks via
//   async global->LDS.  energy never leaves registers.
// =====================================================================
#define EN_ROWS 64
#define EN_NC   32
#define EN_KC   128
#define EN_LD   (EN_KC + 4)   // stride 132 dwords: 16B-aligned, bank-clean

__global__ __launch_bounds__(256) void k_energy(const float* __restrict__ eo,
                                                const float* __restrict__ Wa,
                                                const float* __restrict__ q,
                                                const float* __restrict__ v,
                                                float* __restrict__ scores) {
  __shared__ float sE[EN_ROWS * EN_LD];   // 33 KB
  __shared__ float sW[EN_NC * EN_LD];     // 16.5 KB
  __shared__ float s_part[EN_ROWS];
  const int tid  = threadIdx.x;
  const int lane = tid & 31, w = tid >> 5;
  const int wr = w & 3, wc = w >> 2;      // 4 row strips x 2 col strips of 16
  const int m  = lane & 15, lh = lane >> 4;
  const int r0 = blockIdx.x * EN_ROWS;
  const int b  = r0 / NS;                 // 64 | 2048 -> whole block same batch
  if (tid < EN_ROWS) s_part[tid] = 0.f;

  const uint32_t eBase = lds_off_of(&sE[0]);
  const uint32_t wBase = lds_off_of(&sW[0]);

  float sacc[8];
  #pragma unroll
  for (int i = 0; i < 8; ++i) sacc[i] = 0.f;

  for (int nt = 0; nt < Hh / EN_NC; ++nt) {       // 16 n-tiles
    v8f acc;
    #pragma unroll
    for (int i = 0; i < 8; ++i) acc[i] = 0.f;

    for (int kc = 0; kc < Hh; kc += EN_KC) {      // 4 K-chunks
      __syncthreads();                            // all done reading prev chunk
      // eo chunk: 64 x 128 f32 = 2048 B128 xfers (8 per thread)
      for (int i = tid; i < EN_ROWS * (EN_KC / 4); i += 256) {
        const int row = i >> 5, kq = (i & 31) * 4;
        async_ld_b128(eBase + (uint32_t)(row * EN_LD + kq) * 4,
                      eo + (size_t)(r0 + row) * Hh + kc + kq);
      }
      // Wa_e chunk: 32 k-rows x 128 f32 ; Wa_e[k][h] = Wa[k*2H + H + h]
      for (int i = tid; i < EN_NC * (EN_KC / 4); i += 256) {
        const int row = i >> 5, kq = (i & 31) * 4;
        async_ld_b128(wBase + (uint32_t)(row * EN_LD + kq) * 4,
                      Wa + (size_t)(nt * EN_NC + row) * (2 * Hh) + Hh + kc + kq);
      }
      wait_async0();
      __syncthreads();
      #pragma unroll
      for (int kk = 0; kk < EN_KC; kk += 4) {
        const int ka = kk + lh * 2;
        v2f a;
        a.x = sE[(wr * 16 + m) * EN_LD + ka];
        a.y = sE[(wr * 16 + m) * EN_LD + ka + 1];
        v2f bb;
        bb.x = sW[(wc * 16 + m) * EN_LD + ka];
        bb.y = sW[(wc * 16 + m) * EN_LD + ka + 1];
        acc = wmma_f32(a, bb, acc);
      }
    }
    // fuse: + q[b,k], tanh, * v[k], accumulate per-row partial score
    const int   kcol = nt * EN_NC + wc * 16 + m;
    const float qv = q[b * Hh + kcol];
    const float vv = v[kcol];
    #pragma unroll
    for (int i = 0; i < 8; ++i)
      sacc[i] += tanhf(acc[i] + qv) * vv;
  }
  __syncthreads();
  #pragma unroll
  for (int i = 0; i < 8; ++i)
    atomicAdd(&s_part[wr * 16 + lh * 8 + i], sacc[i]);   // ds_add_f32
  __syncthreads();
  if (tid < EN_ROWS) scores[r0 + tid] = s_part[tid];
}

// =====================================================================
// k_softmax: per batch row of 2048 scores -> weights (exact 2-pass)
// =====================================================================
__global__ __launch_bounds__(256) void k_softmax(const float* __restrict__ scores,
                                                 float* __restrict__ out_w) {
  __shared__ float red[256];
  const int b = blockIdx.x, tid = threadIdx.x;
  const float* s = scores + (size_t)b * NS;
  float mx = -INFINITY;
  for (int i = tid; i < NS; i += 256) mx = fmaxf(mx, s[i]);
  red[tid] = mx; __syncthreads();
  for (int st = 128; st > 0; st >>= 1) {
    if (tid < st) red[tid] = fmaxf(red[tid], red[tid + st]);
    __syncthreads();
  }
  mx = red[0]; __syncthreads();
  float sum = 0.f;
  for (int i = tid; i < NS; i += 256) sum += expf(s[i] - mx);
  red[tid] = sum; __syncthreads();
  for (int st = 128; st > 0; st >>= 1) {
    if (tid < st) red[tid] += red[tid + st];
    __syncthreads();
  }
  const float inv = 1.f / red[0];
  for (int i = tid; i < NS; i += 256)
    out_w[(size_t)b * NS + i] = expf(s[i] - mx) * inv;
}

// =====================================================================
// k_zero + k_context: context[b,h] = sum_s w[b,s] * eo[b*S+s, h]
// =====================================================================
__global__ void k_zero(float* __restrict__ p, int n) {
  const int i = blockIdx.x * blockDim.x + threadIdx.x;
  if (i < n) p[i] = 0.f;
}

__global__ __launch_bounds__(512) void k_context(const float* __restrict__ wgt,
                                                 const float* __restrict__ eo,
                                                 float* __restrict__ ctx) {
  const int b = blockIdx.x, chunk = blockIdx.y;
  const int hcol = threadIdx.x;           // 0..511
  const int s0 = chunk * (NS / 8);        // 8 chunks of 256
  const float* wrow = wgt + (size_t)b * NS;
  float acc = 0.f;
  for (int s = s0; s < s0 + NS / 8; ++s)
    acc += wrow[s] * eo[((size_t)(b * NS + s)) * Hh + hcol];
  atomicAdd(&ctx[b * Hh + hcol], acc);    // global_atomic_add_f32
}

// =====================================================================
extern "C" void kernel_launch(void* const* d_in, const int* in_sizes, int n_in,
                              void* d_out, int out_size, void* d_ws, size_t ws_size,
                              hipStream_t stream) {
  const float* hidden = (const float*)d_in[0];
  const float* enc    = (const float*)d_in[1];
  const float* We     = (const float*)d_in[2];
  const float* be     = (const float*)d_in[3];
  const float* Wh     = (const float*)d_in[4];
  const float* bh     = (const float*)d_in[5];
  const float* Wa     = (const float*)d_in[6];
  const float* ba     = (const float*)d_in[7];
  const float* v      = (const float*)d_in[8];

  float* out     = (float*)d_out;
  float* out_ctx = out;                 // (16, 512)
  float* out_w   = out + NB * Hh;       // (16, 2048)

  float* ws    = (float*)d_ws;          // layout (floats):
  float* ws_h  = ws;                    // [0,      8192)   h
  float* ws_q  = ws + 8192;             // [8192,   16384)  q = h@Wa_h.T + ba
  float* ws_s  = ws + 16384;            // [16384,  49152)  scores
  float* ws_eo = ws + 49152;            // [49152, +16.8M)  eo (64 MB)

  k_h      <<<dim3(NB * Hh / 256), 256, 0, stream>>>(hidden, Wh, bh, ws_h);
  k_q      <<<dim3(NB * Hh / 256), 256, 0, stream>>>(ws_h, Wa, ba, ws_q);
  k_eo     <<<dim3(BSr / EO_ROWS, Hh / EO_COLS), 256, 0, stream>>>(enc, We, be, ws_eo);
  k_energy <<<dim3(BSr / EN_ROWS), 256, 0, stream>>>(ws_eo, Wa, ws_q, v, ws_s);
  k_softmax<<<dim3(NB), 256, 0, stream>>>(ws_s, out_w);
  k_zero   <<<dim3(NB * Hh / 256), 256, 0, stream>>>(out_ctx, NB * Hh);
  k_context<<<dim3(NB, 8), 512, 0, stream>>>(out_w, ws_eo, out_ctx);
}